// GraphConvolution_21698174779868
// MI455X (gfx1250) — compile-verified
//
#include <hip/hip_runtime.h>

typedef __attribute__((ext_vector_type(2))) float v2f;
typedef __attribute__((ext_vector_type(8))) float v8f;
typedef __attribute__((ext_vector_type(4))) float f4v;

#define D 128

// -----------------------------------------------------------------------------
// Phase 1: support = X @ W in full fp32 via V_WMMA_F32_16X16X4_F32.
//
// Workgroup = 256 threads = 8 waves. The WG stages W (128x128 f32 = 64 KB)
// into LDS once, pair-packed so each B fragment is one ds_load_b64:
//     ldsW[(k>>1)*256 + n*2 + (k&1)] = W[k][n]
// Each wave then computes a 16(M) x 128(N) strip, A fragment reused across
// all 8 N-tiles. Load rows are clamped (not guarded) so the inner loop has
// no EXEC manipulation; only the final stores are guarded.
//
// Fragment layouts (ISA 7.12.2, 32-bit operands):
//   A 16x4 : lane L<16 -> {A[M=L][k], A[M=L][k+1]}, lane L+16 -> {k+2, k+3}
//   B 4x16 : lane L<16 -> {B[k][N=L], B[k+1][N=L]}, lane L+16 -> {k+2, k+3}
//   C/D    : VGPR j -> M=j (lanes 0-15), M=j+8 (lanes 16-31), N = lane&15
// -----------------------------------------------------------------------------
__global__ __launch_bounds__(256) void gemm_wmma_f32(const float* __restrict__ X,
                                                     const float* __restrict__ W,
                                                     float* __restrict__ S,
                                                     int n_nodes) {
  __shared__ float ldsW[D * D];          // 64 KB, pair-packed

  const int tid = threadIdx.x;

  // ---- stage W into LDS (once per workgroup) ----
  for (int idx = tid; idx < D * D; idx += 256) {
    const int k = idx >> 7;              // row of W
    const int n = idx & (D - 1);         // col of W
    ldsW[((k >> 1) << 8) + n * 2 + (k & 1)] = W[idx];
  }
  __syncthreads();

  const int lane = tid & 31;
  const int wave = tid >> 5;
  const int l15  = lane & 15;
  const int hi   = lane >> 4;            // 0: lanes 0-15, 1: lanes 16-31
  const int row0 = blockIdx.x * 128 + wave * 16;

  // clamp load row; stores are guarded below
  int m = row0 + l15;
  if (m > n_nodes - 1) m = n_nodes - 1;

  v8f acc[8] = {};                       // 8 N-tile accumulators
  const float* aptr = X + (size_t)m * D + hi * 2;

  for (int k = 0; k < D; k += 4) {
    // A fragment: one global_load_b64, reused across the 8 N-tiles
    const v2f a = *(const v2f*)(aptr + k);
    // pair index for this lane-half: (k + hi*2)/2 = k/2 + hi
    const float* bbase = ldsW + (((k >> 1) + hi) << 8);
#pragma unroll
    for (int t = 0; t < 8; ++t) {
      const int n = t * 16 + l15;
      const v2f b = *(const v2f*)(bbase + n * 2);   // ds_load_b64
      // 8 args: (neg_a, A, neg_b, B, c_mod, C, reuse_a, reuse_b)
      acc[t] = __builtin_amdgcn_wmma_f32_16x16x4_f32(
          false, a, false, b, (short)0, acc[t], false, false);
    }
  }

  // ---- store D: acc[t][j] = D[M = row0 + j + hi*8][N = t*16 + l15] ----
#pragma unroll
  for (int t = 0; t < 8; ++t) {
    const int n = t * 16 + l15;
#pragma unroll
    for (int j = 0; j < 8; ++j) {
      const int mm = row0 + j + hi * 8;
      if (mm < n_nodes) S[(size_t)mm * D + n] = acc[t][j];
    }
  }
}

// -----------------------------------------------------------------------------
// Phase 2a: zero the output (harness poisons d_out with 0xAA).
// -----------------------------------------------------------------------------
__global__ void zero_out_kernel(float* __restrict__ out, long n) {
  long i      = (long)blockIdx.x * blockDim.x + threadIdx.x;
  long stride = (long)gridDim.x * blockDim.x;
  for (; i < n; i += stride) out[i] = 0.0f;
}

// -----------------------------------------------------------------------------
// Phase 2b: edge scatter. One wave per edge: lane handles 4 contiguous
// features (32 lanes * 4 = 128). support/out are L2-resident (51.2 MB each
// vs 192 MB L2), so gathers and fp32 atomics stay on-die.
// -----------------------------------------------------------------------------
__global__ __launch_bounds__(256) void scatter_edges(const float* __restrict__ S,
                                                     const int* __restrict__ rows,
                                                     const int* __restrict__ cols,
                                                     const float* __restrict__ wts,
                                                     float* __restrict__ out,
                                                     int n_edges) {
  long gid = (long)blockIdx.x * blockDim.x + threadIdx.x;
  int  e   = (int)(gid >> 5);     // edge id (uniform per wave)
  int  c   = (int)(gid & 31);     // 4-float chunk within the 128-feature row
  if (e >= n_edges) return;

  const int   r  = rows[e];
  const int   cl = cols[e];
  const float w  = wts[e];

  const f4v s = *(const f4v*)(S + (size_t)cl * D + c * 4);
  float*    o = out + (size_t)r * D + c * 4;

  unsafeAtomicAdd(o + 0, s.x * w);   // global_atomic_add_f32
  unsafeAtomicAdd(o + 1, s.y * w);
  unsafeAtomicAdd(o + 2, s.z * w);
  unsafeAtomicAdd(o + 3, s.w * w);
}

// -----------------------------------------------------------------------------
// Launch
// -----------------------------------------------------------------------------
extern "C" void kernel_launch(void* const* d_in, const int* in_sizes, int n_in,
                              void* d_out, int out_size, void* d_ws, size_t ws_size,
                              hipStream_t stream) {
  const float* X  = (const float*)d_in[0];
  const float* W  = (const float*)d_in[1];
  const int*   ei = (const int*)d_in[2];
  const float* ew = (const float*)d_in[3];
  float*       out     = (float*)d_out;
  float*       support = (float*)d_ws;           // N_NODES*128 fp32 = 51.2 MB

  const int n_nodes = in_sizes[0] / D;
  const int n_edges = in_sizes[3];
  const int* rows = ei;                          // edge_index[0]
  const int* cols = ei + n_edges;                // edge_index[1]

  // GEMM: 8 waves per block, each wave does 16 rows -> 128 rows per block
  const int mblocks = (n_nodes + 127) / 128;
  gemm_wmma_f32<<<mblocks, 256, 0, stream>>>(X, W, support, n_nodes);

  // Zero output
  const long nout = (long)n_nodes * D;
  zero_out_kernel<<<2048, 256, 0, stream>>>(out, nout);

  // Scatter-add: 32 lanes per edge
  const long threads = (long)n_edges * 32;
  const int  blocks  = (int)((threads + 255) / 256);
  scatter_edges<<<blocks, 256, 0, stream>>>(support, rows, cols, ew, out, n_edges);
}